// QuantizedLatent_24026047054740
// MI455X (gfx1250) — compile-verified
//
#include <hip/hip_runtime.h>

typedef __attribute__((ext_vector_type(2))) float    v2f;
typedef __attribute__((ext_vector_type(8))) float    v8f;
typedef __attribute__((ext_vector_type(4))) unsigned v4u;

#define BTOT    8192
#define LDIM    64
#define VDIM    256
#define LBLK    16      // latents per block
#define BBLK    16      // batch rows per block
#define THREADS 256
#define RSTR    20      // reduce-scratch row stride (dwords): 80B, 16B-aligned, 5 coprime 16

static __device__ __forceinline__ unsigned umin2(unsigned a, unsigned b) {
    return a < b ? a : b;
}
static __device__ __forceinline__ unsigned umin3(unsigned a, unsigned b, unsigned c) {
    return umin2(umin2(a, b), c);   // pattern-matches v_min3_u32
}

__global__ __launch_bounds__(THREADS) void vq_wmma_kernel(
    const float* __restrict__ x,
    const float* __restrict__ values,
    float* __restrict__ out)
{
    // B-operand blocks, 32 lanes each: [ll][tt][e]
    //   e in 0..15  -> {-2*c, c^2} for candidate v = tt*16 + e   (rows K0,K1)
    //   e in 16..31 -> {1, 0}                                    (rows K2,K3)
    __shared__ v2f      pairsB[LBLK * 16 * 32];          // 64 KB
    __shared__ float    xT[LBLK * 17];                   // x transposed, padded
    __shared__ float    qlds[BBLK * 17];
    __shared__ unsigned ilds[BBLK * 17];
    __shared__ unsigned red[8 * BBLK * RSTR];            // per-wave argmin scratch

    const int t    = threadIdx.x;
    const int blk  = blockIdx.x;
    const int lgrp = blk & 3;
    const int bsl  = blk >> 2;
    const int b0   = bsl * BBLK;
    const int l0   = lgrp * LBLK;

    // ---- stage codebook into padded B-blocks (coalesced global reads) ----
    for (int i = t; i < LBLK * VDIM; i += THREADS) {
        const int   ll = i >> 8;
        const int   v  = i & 255;
        const float c  = values[(l0 + ll) * VDIM + v];
        v2f p; p.x = -2.0f * c; p.y = c * c;
        pairsB[((ll * 16 + (v >> 4)) * 32) + (v & 15)] = p;
    }
    // constant K2/K3 rows {1,0} for every (ll,tt) block
    for (int i = t; i < LBLK * 16 * 16; i += THREADS) {
        v2f p; p.x = 1.0f; p.y = 0.0f;
        pairsB[((i >> 4) * 32) + 16 + (i & 15)] = p;
    }
    // ---- stage x slab transposed ----
    {
        const int bl = t >> 4, ll = t & 15;
        xT[ll * 17 + bl] = x[(size_t)(b0 + bl) * LDIM + (l0 + ll)];
    }
    __syncthreads();

    const int  wave = t >> 5;
    const int  lane = t & 31;
    const int  nl   = lane & 15;
    const bool lo   = lane < 16;
    unsigned* myred = red + wave * (BBLK * RSTR);

    for (int li = 0; li < 2; ++li) {
        const int   ll = wave * 2 + li;      // local latent 0..15
        const float xm = xT[ll * 17 + nl];

        // A (16x4 f32): lanes 0-15 = [x_m, 1] (K0,K1);
        //               lanes 16-31 = [x_m^2 + bias, 0] (K2,K3)
        v2f a;
        if (lo) { a.x = xm;                      a.y = 1.0f; }
        else    { a.x = xm * xm + 0.0009765625f; a.y = 0.0f; }

        // preload all 16 B operands: one unconditional b64 per tile, all lanes
        const v2f* bbase = pairsB + (ll * 16 * 32) + lane;
        v2f bmv[16];
#pragma unroll
        for (int tt = 0; tt < 16; ++tt) bmv[tt] = bbase[tt * 32];

        unsigned acc[8];
#pragma unroll
        for (int r = 0; r < 8; ++r) acc[r] = 0xFFFFFFFFu;

        v8f cz = {};
#pragma unroll
        for (int tt = 0; tt < 16; tt += 2) {
            // D[m,n] = (x_m - c_n)^2 + bias  (guaranteed >= 0)
            v8f d0 = __builtin_amdgcn_wmma_f32_16x16x4_f32(
                false, a, false, bmv[tt],     (short)0, cz, false, false);
            v8f d1 = __builtin_amdgcn_wmma_f32_16x16x4_f32(
                false, a, false, bmv[tt + 1], (short)0, cz, false, false);
            const unsigned vi0 = (unsigned)(nl | (tt << 4));
            const unsigned vi1 = (unsigned)(nl | ((tt + 1) << 4));
#pragma unroll
            for (int r = 0; r < 8; ++r) {
                const unsigned k0 = (__float_as_uint(d0[r]) & 0xFFFFFF00u) | vi0;
                const unsigned k1 = (__float_as_uint(d1[r]) & 0xFFFFFF00u) | vi1;
                acc[r] = umin3(k0, k1, acc[r]);
            }
        }

        // scatter partial keys: element m = r (+8 for upper half), column nl
        const int mbase = lo ? 0 : 8;
#pragma unroll
        for (int r = 0; r < 8; ++r)
            myred[(mbase + r) * RSTR + nl] = acc[r];
        __threadfence_block();               // order LDS store -> load (same wave)

        // lane nl reduces element nl's 16 partials via 4x ds_load_b128
        const v4u* row = (const v4u*)(myred + nl * RSTR);
        const v4u p0 = row[0], p1 = row[1], p2 = row[2], p3 = row[3];
        unsigned best = umin3(p0.x, p0.y, p0.z);
        best = umin3(best, p0.w, p1.x);
        best = umin3(best, p1.y, p1.z);
        best = umin3(best, p1.w, p2.x);
        best = umin3(best, p2.y, p2.z);
        best = umin3(best, p2.w, p3.x);
        best = umin3(best, p3.y, p3.z);
        best = umin2(best, p3.w);

        const unsigned idx = best & 0xFFu;
        const float c = -0.5f * pairsB[((ll * 16 + (idx >> 4)) * 32) + (idx & 15)].x;
        if (lo) {
            qlds[nl * 17 + ll] = c;
            ilds[nl * 17 + ll] = idx;
        }
    }
    __syncthreads();

    // ---- coalesced 4-plane writeout: thread t -> element (b=t>>4, l=t&15) ----
    {
        const int    bl = t >> 4, ll = t & 15;
        const float  xv = xT[ll * 17 + bl];
        const float  q  = qlds[bl * 17 + ll];
        const float  z  = xv + (q - xv);       // matches reference FP order
        const float  fi = (float)ilds[bl * 17 + ll];
        const size_t BL = (size_t)BTOT * LDIM;
        const size_t g  = (size_t)(b0 + bl) * LDIM + (l0 + ll);
        out[g]          = xv;   // x
        out[BL + g]     = q;    // quantized
        out[2 * BL + g] = z;    // z_hat
        out[3 * BL + g] = fi;   // inds (promoted to f32 by tuple concat)
    }
}

extern "C" void kernel_launch(void* const* d_in, const int* in_sizes, int n_in,
                              void* d_out, int out_size, void* d_ws, size_t ws_size,
                              hipStream_t stream) {
    (void)in_sizes; (void)n_in; (void)d_ws; (void)ws_size; (void)out_size;
    const float* x      = (const float*)d_in[0];
    const float* values = (const float*)d_in[1];
    float*       out    = (float*)d_out;
    const int grid = (BTOT / BBLK) * (LDIM / LBLK);   // 2048 blocks
    vq_wmma_kernel<<<grid, THREADS, 0, stream>>>(x, values, out);
}